// TinyRNN_9191230013556
// MI455X (gfx1250) — compile-verified
//
#include <hip/hip_runtime.h>
#include <math.h>

typedef __attribute__((ext_vector_type(2))) float        v2f;
typedef __attribute__((ext_vector_type(8))) float        v8f;
typedef __attribute__((ext_vector_type(4))) unsigned int v4u;
typedef __attribute__((ext_vector_type(8))) int          v8i;
typedef __attribute__((ext_vector_type(4))) int          v4i;

#define S_LEN 2048
#define B_LEN 4096
#define IN_F 3
#define BLK_B 64                    // batch columns per block (4 waves x 16)
#define T_CHUNK 16                  // timesteps per LDS chunk
#define ROWF (BLK_B * IN_F)         // 192 floats per chunk-row per block
#define NCHUNK (S_LEN / T_CHUNK)    // 128
#define ROW_STRIDE (B_LEN * IN_F)   // 12288 floats between consecutive timesteps

#if defined(__has_builtin)
#if __has_builtin(__builtin_amdgcn_tensor_load_to_lds) && __has_builtin(__builtin_amdgcn_s_wait_tensorcnt)
#define HAVE_TDM 1
#endif
#endif
#ifndef HAVE_TDM
#define HAVE_TDM 0
#endif

__device__ __forceinline__ float fast_tanh(float v) {
#if __has_builtin(__builtin_amdgcn_tanhf)
  return __builtin_amdgcn_tanhf(v);
#elif __has_builtin(__builtin_amdgcn_tanh_f32)
  return __builtin_amdgcn_tanh_f32(v);
#else
  return tanhf(v);
#endif
}

#if HAVE_TDM
// Issue one TDM 2-D tile load: T_CHUNK rows x ROWF floats, row stride ROW_STRIDE
// floats in global memory, packed contiguously into LDS at lds_byte_off.
// Descriptor bitfields per CDNA5 ISA 8.3-8.6 (data_size=4B, type=2, no pad/iterate).
// This toolchain exposes the 6-arg builtin: (v4u, v8i, v4i, v4i, v8i, i32 cpol).
__device__ __forceinline__ void tdm_load_tile(const float* gsrc, unsigned lds_byte_off) {
  unsigned long long ga = (unsigned long long)(uintptr_t)gsrc;
  v4u g0;
  g0.x = 1u;                                               // count=1 valid user descriptor
  g0.y = lds_byte_off;                                     // LDS dest (bytes)
  g0.z = (unsigned)ga;                                     // global_addr[31:0]
  g0.w = (unsigned)((ga >> 32) & 0x01FFFFFFull) | (2u << 30); // addr[56:32] | type=2
  v8i g1;
  g1[0] = (int)(2u << 16);                                 // mask=0, data_size=2 (4B)
  g1[1] = (int)((unsigned)(ROWF & 0xFFFF) << 16);          // tensor_dim0 [15:0]
  g1[2] = (int)((unsigned)(ROWF >> 16) |
                ((unsigned)(T_CHUNK & 0xFFFF) << 16));     // dim0 hi | tensor_dim1 lo
  g1[3] = (int)((unsigned)(T_CHUNK >> 16) |
                ((unsigned)ROWF << 16));                   // dim1 hi | tile_dim0
  g1[4] = (int)(T_CHUNK & 0xFFFF);                         // tile_dim1 | tile_dim2=0
  g1[5] = (int)ROW_STRIDE;                                 // tensor_dim0_stride[31:0]
  g1[6] = 0;                                               // stride hi | dim1_stride lo
  g1[7] = 0;
  v4i z4 = {0, 0, 0, 0};                                   // groups 2/3 unused (2-D)
  v8i z8 = {0, 0, 0, 0, 0, 0, 0, 0};
  __builtin_amdgcn_tensor_load_to_lds(g0, g1, z4, z4, z8, 0);
}
#endif

__global__ void __launch_bounds__(BLK_B * 2 /*128 threads*/)
tiny_rnn_scan(const float* __restrict__ x,
              const float* __restrict__ Wih,
              const float* __restrict__ Whh,
              float* __restrict__ out) {
  __shared__ float sbuf[2 * T_CHUNK * ROWF];               // 24 KB double buffer

  const int tid  = threadIdx.x;
  const int lane = tid & 31;
  const int wave = tid >> 5;                               // 0..3
  const int n    = lane & 15;                              // column within wave tile
  const int kh   = lane >> 4;                              // K-half for A/B layouts
  const int blk  = blockIdx.x;
  const int bcol = blk * BLK_B + wave * 16 + n;            // batch column (lanes 0-15)

  // Recurrent 2x2 weights (uniform -> scalar loads).
  const float w00 = Whh[0], w01 = Whh[1], w10 = Whh[2], w11 = Whh[3];

  // A = W_ih zero-padded to 16x4 (M=h rows, K=input). ISA 16x4 f32 A layout:
  // lanes 0-15 hold K=0 (a.x), K=1 (a.y); lanes 16-31 hold K=2 (a.x), K=3 (a.y).
  v2f a; a.x = 0.0f; a.y = 0.0f;
  if (n < 2) {
    a.x = kh ? Wih[n * 3 + 2] : Wih[n * 3 + 0];
    a.y = kh ? 0.0f           : Wih[n * 3 + 1];
  }

  const float* gblk = x + (size_t)blk * BLK_B * IN_F;      // this block's column slice
  const unsigned bufBytes = (unsigned)(T_CHUNK * ROWF * 4);
  const unsigned lds_base = (unsigned)(uintptr_t)sbuf;     // flat LDS addr[31:0] = LDS offset

#if HAVE_TDM
  if (wave == 0) tdm_load_tile(gblk, lds_base);            // prefetch chunk 0
#endif

  float h0 = 0.0f, h1 = 0.0f;
  const int col3 = (wave * 16 + n) * 3;

  for (int c = 0; c < NCHUNK; ++c) {
    const int cur = c & 1;
#if HAVE_TDM
    if (wave == 0) {
      if (c + 1 < NCHUNK) {
        tdm_load_tile(gblk + (size_t)(c + 1) * T_CHUNK * ROW_STRIDE,
                      lds_base + (unsigned)((c + 1) & 1) * bufBytes);
        __builtin_amdgcn_s_wait_tensorcnt(1);              // chunk c landed; c+1 in flight
      } else {
        __builtin_amdgcn_s_wait_tensorcnt(0);
      }
    }
    __syncthreads();
#else
    {   // fallback: cooperative synchronous chunk load
      const float* gsrc = gblk + (size_t)c * T_CHUNK * ROW_STRIDE;
      float* dst = sbuf + cur * (T_CHUNK * ROWF);
      for (int i = tid; i < T_CHUNK * ROWF; i += (int)blockDim.x) {
        int row = i / ROWF, j = i - row * ROWF;
        dst[i] = gsrc[(size_t)row * ROW_STRIDE + j];
      }
      __syncthreads();
    }
#endif
    const float* lbuf = sbuf + cur * (T_CHUNK * ROWF);

#pragma unroll 4
    for (int rrow = 0; rrow < T_CHUNK; ++rrow) {
      const float* lr = lbuf + rrow * ROWF;
      // B = x-slice as 3x16 (zero-padded 4x16). ISA B layout: vgpr0 lanes0-15 K=0,
      // lanes16-31 K=1; vgpr1 lanes0-15 K=2, lanes16-31 K=3(=0).
      v2f b;
      b.x = lr[col3 + kh];
      b.y = kh ? 0.0f : lr[col3 + 2];

      v8f acc = {0.0f, 0.0f, 0.0f, 0.0f, 0.0f, 0.0f, 0.0f, 0.0f};
      // xp tile: D[m][n] = sum_k W_ih[m][k] * x[s][b_n][k]; rows 0/1 -> acc[0]/acc[1]
      acc = __builtin_amdgcn_wmma_f32_16x16x4_f32(
          /*neg_a=*/false, a, /*neg_b=*/false, b,
          /*c_mod=*/(short)0, acc, /*reuse_a=*/false, /*reuse_b=*/false);

      // Serial critical path (VALU only): 2 FMAs + tanh; WMMA stays off-chain.
      const float pre0 = acc[0] + w00 * h0 + w01 * h1;
      const float pre1 = acc[1] + w10 * h0 + w11 * h1;
      h0 = fast_tanh(pre0);
      h1 = fast_tanh(pre1);

      if (lane < 16) {                                     // lanes 16-31 hold pad rows
        const int s = c * T_CHUNK + rrow;
        float2 o; o.x = h0; o.y = h1;
        *(float2*)(out + ((size_t)s * B_LEN + bcol) * 2) = o;
      }
    }
    __syncthreads();   // everyone done with lbuf before TDM reuses this buffer
  }
}

extern "C" void kernel_launch(void* const* d_in, const int* in_sizes, int n_in,
                              void* d_out, int out_size, void* d_ws, size_t ws_size,
                              hipStream_t stream) {
  (void)in_sizes; (void)n_in; (void)out_size; (void)d_ws; (void)ws_size;
  const float* x   = (const float*)d_in[0];
  const float* Wih = (const float*)d_in[1];
  const float* Whh = (const float*)d_in[2];
  float* out = (float*)d_out;
  dim3 grid(B_LEN / BLK_B);   // 64 blocks
  dim3 block(BLK_B * 2);      // 128 threads = 4 wave32s, 16 batch columns each
  hipLaunchKernelGGL(tiny_rnn_scan, grid, block, 0, stream, x, Wih, Whh, out);
}